// DarkManifoldV2_74663711474150
// MI455X (gfx1250) — compile-verified
//
#include <hip/hip_runtime.h>
#include <math.h>

// Problem constants (from reference): B=16, N=1024, H=64, SD=64, MINS=2
#define BDIM  16
#define NDIM  1024
#define HDIM  64
#define SDDIM 64
#define NB    64           // GJ block size
#define NBLK  (NDIM / NB)  // 16 block steps
#define ETA_C 0.01f

typedef __attribute__((ext_vector_type(2))) float v2f;
typedef __attribute__((ext_vector_type(8))) float v8f;

// ---------------------------------------------------------------------------
// WMMA helpers: V_WMMA_F32_16X16X4_F32 fragment layouts (CDNA5 ISA 7.12.2)
//   A (16x4):  lanes 0-15 -> M=lane, VGPR0=K0,VGPR1=K1 ; lanes 16-31 -> K2,K3
//   B (4x16):  lanes hold N=lane&15 ; K = vgpr + 2*(lane>=16)
//   C (16x16): VGPR r: M = r + 8*(lane>=16), N = lane&15
// ---------------------------------------------------------------------------
__device__ __forceinline__ v2f load_afrag(const float* A, int lda, int m0, int kk) {
  int lane = threadIdx.x & 31;
  const float* p = A + (size_t)(m0 + (lane & 15)) * lda + kk + 2 * (lane >> 4);
  v2f r; r.x = p[0]; r.y = p[1];
  return r;
}
__device__ __forceinline__ v2f load_bfrag(const float* Bm, int ldb, int n0, int kk) {
  int lane = threadIdx.x & 31;
  int krow = kk + 2 * (lane >> 4);
  int n = n0 + (lane & 15);
  v2f r;
  r.x = Bm[(size_t)krow * ldb + n];
  r.y = Bm[(size_t)(krow + 1) * ldb + n];
  return r;
}
__device__ __forceinline__ v8f load_cfrag(const float* C, int ldc, int m0, int n0) {
  int lane = threadIdx.x & 31;
  int mh = m0 + 8 * (lane >> 4), n = n0 + (lane & 15);
  v8f c;
#pragma unroll
  for (int r = 0; r < 8; ++r) c[r] = C[(size_t)(mh + r) * ldc + n];
  return c;
}
__device__ __forceinline__ void store_cfrag(float* C, int ldc, int m0, int n0, v8f c) {
  int lane = threadIdx.x & 31;
  int mh = m0 + 8 * (lane >> 4), n = n0 + (lane & 15);
#pragma unroll
  for (int r = 0; r < 8; ++r) C[(size_t)(mh + r) * ldc + n] = c[r];
}

// f32 WMMA: ISA requires NEG[0]/NEG[1]==0 for float types (only C-neg exists),
// so complex-arithmetic signs are applied by negating fragments in VALU (cheap
// v_xor, co-issues with the matrix pipe).
__device__ __forceinline__ v8f wmma4(v2f a, v2f b, v8f c) {
  return __builtin_amdgcn_wmma_f32_16x16x4_f32(false, a, false, b, (short)0, c,
                                               false, false);
}

// Complex MAC via 4 real WMMAs; template bools = sign of each term:
//   cr += (NRR? -:+) ar*br + (NII? -:+) ai*bi
//   ci += (NRI? -:+) ar*bi + (NIR? -:+) ai*br
template <bool NRR, bool NII, bool NRI, bool NIR>
__device__ __forceinline__ void cwmma(v2f ar, v2f ai, v2f br, v2f bi, v8f& cr, v8f& ci) {
  v2f arn = -ar, ain = -ai;  // dead-code eliminated when unused
  cr = wmma4(NRR ? arn : ar, br, cr);
  cr = wmma4(NII ? ain : ai, bi, cr);
  ci = wmma4(NRI ? arn : ar, bi, ci);
  ci = wmma4(NIR ? ain : ai, br, ci);
}

// 8 waves per block; wave owns two 16x16 tiles of a 64x64 block (shared A frag)
__device__ __forceinline__ void wave_tiles(int& m0, int& n0) {
  int wave = threadIdx.x >> 5;
  m0 = (wave >> 1) * 16;
  n0 = (wave & 1) * 32;
}

// ---------------------------------------------------------------------------
// Small kernels: state_probs, MLPs + decoherence, omega mean
// ---------------------------------------------------------------------------
__global__ void cell_probs_kernel(const float* __restrict__ gene,
                                  const float* __restrict__ semb,
                                  float* __restrict__ out_sp) {
  int b = blockIdx.x;
  int h = threadIdx.x;  // 64 threads
  float s = 0.f;
  const float* g = gene + (size_t)b * NDIM * HDIM + h;
  for (int n = 0; n < NDIM; ++n) s += g[(size_t)n * HDIM];
  float cell = s * (1.f / NDIM);
  __shared__ float shA[HDIM], shB[HDIM];
  shA[h] = cell * cell;
  __syncthreads();
  for (int off = 32; off > 0; off >>= 1) { if (h < off) shA[h] += shA[h + off]; __syncthreads(); }
  float cn = cell / fmaxf(sqrtf(shA[0]), 1e-12f);
  __syncthreads();
  float logit[2];
#pragma unroll
  for (int si = 0; si < 2; ++si) {
    float e = semb[si * SDDIM + h];
    shA[h] = e * e; shB[h] = cn * e;
    __syncthreads();
    for (int off = 32; off > 0; off >>= 1) {
      if (h < off) { shA[h] += shA[h + off]; shB[h] += shB[h + off]; }
      __syncthreads();
    }
    logit[si] = (shB[0] / fmaxf(sqrtf(shA[0]), 1e-12f)) * 10.f;  // /0.1 temp
    __syncthreads();
  }
  if (h == 0) {
    float m = fmaxf(logit[0], logit[1]);
    float e0 = expf(logit[0] - m), e1 = expf(logit[1] - m);
    float inv = 1.f / (e0 + e1);
    out_sp[b * 2 + 0] = e0 * inv;
    out_sp[b * 2 + 1] = e1 * inv;
  }
}

// One wave32 per (b,n) row: both 64->32->1 MLPs, gamma, decohered row, omega.
__global__ void mlp_decohere_kernel(const float* __restrict__ gene,
                                    const float* __restrict__ coh,
                                    const float* __restrict__ gbase,
                                    const float* __restrict__ ew1, const float* __restrict__ eb1,
                                    const float* __restrict__ ew2, const float* __restrict__ eb2,
                                    const float* __restrict__ ow1, const float* __restrict__ ob1,
                                    const float* __restrict__ ow2, const float* __restrict__ ob2,
                                    float* __restrict__ out_deco,
                                    float* __restrict__ out_gamma,
                                    float* __restrict__ ws_omega) {
  int wave = threadIdx.x >> 5;
  int lane = threadIdx.x & 31;
  size_t row = (size_t)blockIdx.x * 8 + wave;  // < B*N
  int n = (int)(row & (NDIM - 1));
  __shared__ float xs[8][HDIM];
  const float* x = gene + row * HDIM;
  xs[wave][lane]      = x[lane];
  xs[wave][lane + 32] = x[lane + 32];
  __syncthreads();
  float he = eb1[lane], ho = ob1[lane];
#pragma unroll 8
  for (int t = 0; t < HDIM; ++t) {
    float xv = xs[wave][t];
    he += xv * ew1[t * 32 + lane];
    ho += xv * ow1[t * 32 + lane];
  }
  he = he / (1.f + expf(-he));  // silu
  ho = ho / (1.f + expf(-ho));
  float pe = he * ew2[lane];
  float po = ho * ow2[lane];
  float cs = xs[wave][lane] + xs[wave][lane + 32];
#pragma unroll
  for (int off = 16; off > 0; off >>= 1) {
    pe += __shfl_xor(pe, off, 32);
    po += __shfl_xor(po, off, 32);
    cs += __shfl_xor(cs, off, 32);
  }
  float env = 1.f / (1.f + expf(-(pe + eb2[0])));
  float omega = po + ob2[0];
  float gamma = (1.f / (1.f + expf(-gbase[n]))) * (1.f + env);
  float classical = cs * (1.f / HDIM);
  const float* c = coh + row * HDIM;
  float* d = out_deco + row * HDIM;
  d[lane]      = (1.f - gamma) * c[lane]      + gamma * classical;
  d[lane + 32] = (1.f - gamma) * c[lane + 32] + gamma * classical;
  if (lane == 0) { out_gamma[row] = gamma; ws_omega[row] = omega; }
}

__global__ void omega_mean_kernel(const float* __restrict__ ws_omega,
                                  float* __restrict__ ws_omega_mean) {
  int b = blockIdx.x, t = threadIdx.x;
  float s = 0.f;
  for (int n = t; n < NDIM; n += 256) s += ws_omega[(size_t)b * NDIM + n];
  __shared__ float sh[256];
  sh[t] = s; __syncthreads();
  for (int off = 128; off > 0; off >>= 1) { if (t < off) sh[t] += sh[t + off]; __syncthreads(); }
  if (t == 0) ws_omega_mean[b] = sh[0] * (1.f / NDIM);
}

// ---------------------------------------------------------------------------
// Propagator: R = (omega_b + i*eta) I - H_sym ; in-place blocked Gauss-Jordan.
// Re plane lives in d_out's propagator slab; Im plane + scratch in d_ws.
// slot layout (floats): [N*N Im][NB*NB pinv_re][NB*NB pinv_im][NB*N prow_re][NB*N prow_im]
// ---------------------------------------------------------------------------
__global__ void init_R_kernel(const float* __restrict__ Hmat,
                              const float* __restrict__ omega_mean, int b0,
                              float* __restrict__ prop, float* __restrict__ ws_mat,
                              size_t slotElems) {
  int s = blockIdx.z, b = b0 + s;
  float* Re = prop + (size_t)b * NDIM * NDIM;
  float* Im = ws_mat + (size_t)s * slotElems;
  size_t idx = (size_t)blockIdx.x * 256 + threadIdx.x;
  int i = (int)(idx >> 10), j = (int)(idx & (NDIM - 1));
  float hs = 0.5f * (Hmat[idx] + Hmat[(size_t)j * NDIM + i]);
  float diag = (i == j) ? 1.f : 0.f;
  Re[idx] = omega_mean[b] * diag - hs;
  Im[idx] = ETA_C * diag;
}

// Unblocked complex GJ of the 64x64 pivot block in LDS (no pivoting: matrix is
// a complex shift of a real symmetric matrix -> all leading minors nonsingular)
__global__ void invert_pivot_kernel(float* __restrict__ prop, float* __restrict__ ws_mat,
                                    size_t slotElems, int b0, int k) {
  int s = blockIdx.z, b = b0 + s;
  float* Re = prop + (size_t)b * NDIM * NDIM;
  float* Im = ws_mat + (size_t)s * slotElems;
  float* pvr = Im + (size_t)NDIM * NDIM;
  float* pvi = pvr + NB * NB;
  __shared__ float Pr[NB][NB + 1], Pi[NB][NB + 1];
  __shared__ float colr[NB], coli[NB];
  int j = threadIdx.x;  // 64 threads, thread j owns column j
  size_t base = (size_t)(k * NB) * NDIM + (size_t)k * NB;
  for (int t = 0; t < NB; ++t) {
    Pr[j][t] = Re[base + (size_t)j * NDIM + t];
    Pi[j][t] = Im[base + (size_t)j * NDIM + t];
  }
  __syncthreads();
  for (int kk = 0; kk < NB; ++kk) {
    colr[j] = Pr[j][kk]; coli[j] = Pi[j][kk];
    __syncthreads();
    float dr = Pr[kk][kk], di = Pi[kk][kk];
    float den = 1.f / (dr * dr + di * di);
    float ipr = dr * den, ipi = -di * den;
    float rkr, rki;
    if (j == kk) { rkr = ipr; rki = ipi; }
    else { float ar = Pr[kk][j], ai = Pi[kk][j];
           rkr = ar * ipr - ai * ipi; rki = ar * ipi + ai * ipr; }
    __syncthreads();
    Pr[kk][j] = rkr; Pi[kk][j] = rki;
    for (int i = 0; i < NB; ++i) {
      if (i == kk) continue;
      float tr = colr[i], ti = coli[i];
      if (j == kk) {
        Pr[i][j] = -(tr * ipr - ti * ipi);
        Pi[i][j] = -(tr * ipi + ti * ipr);
      } else {
        Pr[i][j] -= tr * rkr - ti * rki;
        Pi[i][j] -= tr * rki + ti * rkr;
      }
    }
    __syncthreads();
  }
  for (int t = 0; t < NB; ++t) {
    Re[base + (size_t)j * NDIM + t] = Pr[j][t];
    Im[base + (size_t)j * NDIM + t] = Pi[j][t];
    pvr[j * NB + t] = Pr[j][t];
    pvi[j * NB + t] = Pi[j][t];
  }
}

// prow[jb] = Pinv * M[k][jb]   (jb != k)    — 4 fp32 WMMAs per tile per K-step
__global__ void prow_kernel(float* __restrict__ prop, float* __restrict__ ws_mat,
                            size_t slotElems, int b0, int k) {
  int s = blockIdx.z, b = b0 + s;
  const float* Re = prop + (size_t)b * NDIM * NDIM;
  const float* Im = ws_mat + (size_t)s * slotElems;
  const float* pvr = Im + (size_t)NDIM * NDIM;
  const float* pvi = pvr + NB * NB;
  float* prr = (float*)(pvi + NB * NB);
  float* pri = prr + (size_t)NB * NDIM;
  int jb = blockIdx.x; jb += (jb >= k);
  const float* Bre = Re + (size_t)(k * NB) * NDIM + (size_t)jb * NB;
  const float* Bim = Im + (size_t)(k * NB) * NDIM + (size_t)jb * NB;
  int m0, n0; wave_tiles(m0, n0);
  v8f cr0 = {0.f,0.f,0.f,0.f,0.f,0.f,0.f,0.f}, ci0 = cr0, cr1 = cr0, ci1 = cr0;
#pragma unroll 4
  for (int kk = 0; kk < NB; kk += 4) {
    v2f ar = load_afrag(pvr, NB, m0, kk);
    v2f ai = load_afrag(pvi, NB, m0, kk);
    v2f br0 = load_bfrag(Bre, NDIM, n0, kk),      bi0 = load_bfrag(Bim, NDIM, n0, kk);
    v2f br1 = load_bfrag(Bre, NDIM, n0 + 16, kk), bi1 = load_bfrag(Bim, NDIM, n0 + 16, kk);
    cwmma<false, true, false, false>(ar, ai, br0, bi0, cr0, ci0);  // + P*B (complex)
    cwmma<false, true, false, false>(ar, ai, br1, bi1, cr1, ci1);
  }
  store_cfrag(prr + (size_t)jb * NB, NDIM, m0, n0, cr0);
  store_cfrag(prr + (size_t)jb * NB, NDIM, m0, n0 + 16, cr1);
  store_cfrag(pri + (size_t)jb * NB, NDIM, m0, n0, ci0);
  store_cfrag(pri + (size_t)jb * NB, NDIM, m0, n0 + 16, ci1);
}

// M[i][j] -= M[i][k] * prow[j]   (i,j != k): 225 block-GEMMs, accumulate in C
__global__ void trailing_kernel(float* __restrict__ prop, float* __restrict__ ws_mat,
                                size_t slotElems, int b0, int k) {
  int s = blockIdx.z, b = b0 + s;
  float* Re = prop + (size_t)b * NDIM * NDIM;
  float* Im = ws_mat + (size_t)s * slotElems;
  const float* prr = Im + (size_t)NDIM * NDIM + 2 * NB * NB;
  const float* pri = prr + (size_t)NB * NDIM;
  int ii = blockIdx.x / (NBLK - 1), jj = blockIdx.x % (NBLK - 1);
  ii += (ii >= k); jj += (jj >= k);
  const float* Are = Re + (size_t)(ii * NB) * NDIM + (size_t)k * NB;
  const float* Aim = Im + (size_t)(ii * NB) * NDIM + (size_t)k * NB;
  const float* Bre = prr + (size_t)jj * NB;
  const float* Bim = pri + (size_t)jj * NB;
  float* Cre = Re + (size_t)(ii * NB) * NDIM + (size_t)jj * NB;
  float* Cim = Im + (size_t)(ii * NB) * NDIM + (size_t)jj * NB;
  int m0, n0; wave_tiles(m0, n0);
  v8f cr0 = load_cfrag(Cre, NDIM, m0, n0),      ci0 = load_cfrag(Cim, NDIM, m0, n0);
  v8f cr1 = load_cfrag(Cre, NDIM, m0, n0 + 16), ci1 = load_cfrag(Cim, NDIM, m0, n0 + 16);
#pragma unroll 4
  for (int kk = 0; kk < NB; kk += 4) {
    v2f ar = load_afrag(Are, NDIM, m0, kk);
    v2f ai = load_afrag(Aim, NDIM, m0, kk);
    v2f br0 = load_bfrag(Bre, NDIM, n0, kk),      bi0 = load_bfrag(Bim, NDIM, n0, kk);
    v2f br1 = load_bfrag(Bre, NDIM, n0 + 16, kk), bi1 = load_bfrag(Bim, NDIM, n0 + 16, kk);
    cwmma<true, false, true, true>(ar, ai, br0, bi0, cr0, ci0);  // C -= A*B (complex)
    cwmma<true, false, true, true>(ar, ai, br1, bi1, cr1, ci1);
  }
  store_cfrag(Cre, NDIM, m0, n0, cr0);      store_cfrag(Cim, NDIM, m0, n0, ci0);
  store_cfrag(Cre, NDIM, m0, n0 + 16, cr1); store_cfrag(Cim, NDIM, m0, n0 + 16, ci1);
}

// M[i][k] = -(M[i][k] * Pinv)  (i != k): stage T in LDS to break in-place hazard
__global__ void colupdate_kernel(float* __restrict__ prop, float* __restrict__ ws_mat,
                                 size_t slotElems, int b0, int k) {
  int s = blockIdx.z, b = b0 + s;
  float* Re = prop + (size_t)b * NDIM * NDIM;
  float* Im = ws_mat + (size_t)s * slotElems;
  const float* pvr = Im + (size_t)NDIM * NDIM;
  const float* pvi = pvr + NB * NB;
  int ii = blockIdx.x; ii += (ii >= k);
  float* Tre = Re + (size_t)(ii * NB) * NDIM + (size_t)k * NB;
  float* Tim = Im + (size_t)(ii * NB) * NDIM + (size_t)k * NB;
  __shared__ float Lr[NB * NB], Li[NB * NB];
  for (int t = threadIdx.x; t < NB * NB; t += 256) {
    Lr[t] = Tre[(size_t)(t >> 6) * NDIM + (t & 63)];
    Li[t] = Tim[(size_t)(t >> 6) * NDIM + (t & 63)];
  }
  __syncthreads();
  int m0, n0; wave_tiles(m0, n0);
  v8f cr0 = {0.f,0.f,0.f,0.f,0.f,0.f,0.f,0.f}, ci0 = cr0, cr1 = cr0, ci1 = cr0;
#pragma unroll 4
  for (int kk = 0; kk < NB; kk += 4) {
    v2f ar = load_afrag(Lr, NB, m0, kk);
    v2f ai = load_afrag(Li, NB, m0, kk);
    v2f br0 = load_bfrag(pvr, NB, n0, kk),      bi0 = load_bfrag(pvi, NB, n0, kk);
    v2f br1 = load_bfrag(pvr, NB, n0 + 16, kk), bi1 = load_bfrag(pvi, NB, n0 + 16, kk);
    cwmma<true, false, true, true>(ar, ai, br0, bi0, cr0, ci0);  // -T*Pinv
    cwmma<true, false, true, true>(ar, ai, br1, bi1, cr1, ci1);
  }
  store_cfrag(Tre, NDIM, m0, n0, cr0);      store_cfrag(Tim, NDIM, m0, n0, ci0);
  store_cfrag(Tre, NDIM, m0, n0 + 16, cr1); store_cfrag(Tim, NDIM, m0, n0 + 16, ci1);
}

// M[k][j] = prow[j]  (j != k)
__global__ void copy_prow_kernel(float* __restrict__ prop, float* __restrict__ ws_mat,
                                 size_t slotElems, int b0, int k) {
  int s = blockIdx.z, b = b0 + s;
  float* Re = prop + (size_t)b * NDIM * NDIM;
  float* Im = ws_mat + (size_t)s * slotElems;
  const float* prr = Im + (size_t)NDIM * NDIM + 2 * NB * NB;
  const float* pri = prr + (size_t)NB * NDIM;
  int idx = blockIdx.x * 256 + threadIdx.x;  // < NB*NDIM
  int r = idx >> 10, c = idx & (NDIM - 1);
  if ((c >> 6) == k) return;
  Re[(size_t)(k * NB + r) * NDIM + c] = prr[(size_t)r * NDIM + c];
  Im[(size_t)(k * NB + r) * NDIM + c] = pri[(size_t)r * NDIM + c];
}

__global__ void abs_kernel(float* __restrict__ prop, const float* __restrict__ ws_mat,
                           size_t slotElems, int b0) {
  int s = blockIdx.z, b = b0 + s;
  float* Re = prop + (size_t)b * NDIM * NDIM;
  const float* Im = ws_mat + (size_t)s * slotElems;
  size_t idx = (size_t)blockIdx.x * 256 + threadIdx.x;
  float re = Re[idx], im = Im[idx];
  Re[idx] = sqrtf(re * re + im * im);
}

// ---------------------------------------------------------------------------
extern "C" void kernel_launch(void* const* d_in, const int* in_sizes, int n_in,
                              void* d_out, int out_size, void* d_ws, size_t ws_size,
                              hipStream_t stream) {
  (void)in_sizes; (void)n_in; (void)out_size;
  const float* gene  = (const float*)d_in[0];
  const float* coh   = (const float*)d_in[1];
  const float* semb  = (const float*)d_in[2];
  const float* gbase = (const float*)d_in[3];
  const float* ew1 = (const float*)d_in[4];
  const float* eb1 = (const float*)d_in[5];
  const float* ew2 = (const float*)d_in[6];
  const float* eb2 = (const float*)d_in[7];
  const float* ow1 = (const float*)d_in[8];
  const float* ob1 = (const float*)d_in[9];
  const float* ow2 = (const float*)d_in[10];
  const float* ob2 = (const float*)d_in[11];
  const float* Hmat = (const float*)d_in[12];

  float* out = (float*)d_out;
  float* out_sp    = out;                                  // B*MINS = 32
  float* out_deco  = out + 32;                             // B*N*H
  float* out_gamma = out_deco + (size_t)BDIM * NDIM * HDIM;
  float* out_prop  = out_gamma + (size_t)BDIM * NDIM;      // B*N*N (also Re scratch)

  float* ws = (float*)d_ws;
  float* ws_omega      = ws;                               // B*N
  float* ws_omega_mean = ws_omega + (size_t)BDIM * NDIM;   // B
  size_t smallElems = ((size_t)BDIM * NDIM + BDIM + 63) & ~(size_t)63;
  float* ws_mat = ws + smallElems;

  const size_t slotElems = (size_t)NDIM * NDIM + 2 * NB * NB + 2 * (size_t)NB * NDIM;
  size_t avail = (ws_size / 4 > smallElems) ? (ws_size / 4 - smallElems) : 0;
  int G = (int)(avail / slotElems);
  if (G < 1) G = 1;
  if (G > BDIM) G = BDIM;

  cell_probs_kernel<<<BDIM, HDIM, 0, stream>>>(gene, semb, out_sp);
  mlp_decohere_kernel<<<(BDIM * NDIM) / 8, 256, 0, stream>>>(
      gene, coh, gbase, ew1, eb1, ew2, eb2, ow1, ob1, ow2, ob2,
      out_deco, out_gamma, ws_omega);
  omega_mean_kernel<<<BDIM, 256, 0, stream>>>(ws_omega, ws_omega_mean);

  for (int b0 = 0; b0 < BDIM; b0 += G) {
    int g = (BDIM - b0 < G) ? (BDIM - b0) : G;
    init_R_kernel<<<dim3(NDIM * NDIM / 256, 1, g), 256, 0, stream>>>(
        Hmat, ws_omega_mean, b0, out_prop, ws_mat, slotElems);
    for (int k = 0; k < NBLK; ++k) {
      invert_pivot_kernel<<<dim3(1, 1, g), NB, 0, stream>>>(out_prop, ws_mat, slotElems, b0, k);
      prow_kernel<<<dim3(NBLK - 1, 1, g), 256, 0, stream>>>(out_prop, ws_mat, slotElems, b0, k);
      trailing_kernel<<<dim3((NBLK - 1) * (NBLK - 1), 1, g), 256, 0, stream>>>(
          out_prop, ws_mat, slotElems, b0, k);
      colupdate_kernel<<<dim3(NBLK - 1, 1, g), 256, 0, stream>>>(out_prop, ws_mat, slotElems, b0, k);
      copy_prow_kernel<<<dim3(NB * NDIM / 256, 1, g), 256, 0, stream>>>(
          out_prop, ws_mat, slotElems, b0, k);
    }
    abs_kernel<<<dim3(NDIM * NDIM / 256, 1, g), 256, 0, stream>>>(out_prop, ws_mat, slotElems, b0);
  }
}